// MPNN_GNN3D_L_K_U_86878598464025
// MI455X (gfx1250) — compile-verified
//
#include <hip/hip_runtime.h>
#include <math.h>

#define EPS 1e-5f

constexpr int Br = 64;        // BATCH * NBR
constexpr int C  = 64;        // EMB
constexpr int L  = 64;
constexpr int Kd = 32;
constexpr int U  = 8;
constexpr int KU = Kd * U;    // 256
constexpr int LK = L * Kd;    // 2048
constexpr int LU = L * U;     // 512
constexpr int S  = L * Kd * U;// 16384

typedef float v2f __attribute__((ext_vector_type(2)));
typedef float v8f __attribute__((ext_vector_type(8)));

// ---------------------------------------------------------------- utilities
__global__ void kZeroStats(float* st) { st[threadIdx.x] = 0.f; }

// ---------------------------------------------------------------- H means
__global__ __launch_bounds__(256) void kHMeanU(const float* __restrict__ re,
                                               const float* __restrict__ im,
                                               float* __restrict__ out) {
  int tid = blockIdx.x * 256 + threadIdx.x;          // Br*2*LK
  int p = tid % LK; int c = (tid / LK) & 1; int b = tid / (2 * LK);
  const float* src = (c ? im : re) + (size_t)b * S + (size_t)p * U;
  float s = 0.f;
#pragma unroll
  for (int u = 0; u < U; u++) s += src[u];
  out[tid] = s * (1.f / U);
}

__global__ __launch_bounds__(256) void kHMeanK(const float* __restrict__ re,
                                               const float* __restrict__ im,
                                               float* __restrict__ out) {
  int tid = blockIdx.x * 256 + threadIdx.x;          // Br*2*LU
  int p = tid % LU; int c = (tid / LU) & 1; int b = tid / (2 * LU);
  int l = p / U, u = p % U;
  const float* src = (c ? im : re) + (size_t)b * S + (size_t)l * KU + u;
  float s = 0.f;
#pragma unroll 8
  for (int k = 0; k < Kd; k++) s += src[(size_t)k * U];
  out[tid] = s * (1.f / Kd);
}

__global__ __launch_bounds__(256) void kHMeanL(const float* __restrict__ re,
                                               const float* __restrict__ im,
                                               float* __restrict__ out) {
  int tid = blockIdx.x * 256 + threadIdx.x;          // Br*2*KU
  int r = tid % KU; int c = (tid / KU) & 1; int b = tid / (2 * KU);
  const float* src = (c ? im : re) + (size_t)b * S + r;
  float s = 0.f;
#pragma unroll 8
  for (int l = 0; l < L; l++) s += src[(size_t)l * KU];
  out[tid] = s * (1.f / L);
}

// ---------------------------------------------------------------- A means (64ch)
__global__ __launch_bounds__(256) void kAMeanU(const float* __restrict__ A,
                                               float* __restrict__ out) {
  int tid = blockIdx.x * 256 + threadIdx.x;          // Br*C*LK
  int p = tid % LK; size_t bc = tid / LK;
  const float* src = A + bc * S + (size_t)p * U;
  float s = 0.f;
#pragma unroll
  for (int u = 0; u < U; u++) s += src[u];
  out[tid] = s * (1.f / U);
}

__global__ __launch_bounds__(256) void kAMeanK(const float* __restrict__ A,
                                               float* __restrict__ out) {
  int tid = blockIdx.x * 256 + threadIdx.x;          // Br*C*LU
  int p = tid % LU; size_t bc = tid / LU;
  int l = p / U, u = p % U;
  const float* src = A + bc * S + (size_t)l * KU + u;
  float s = 0.f;
#pragma unroll 8
  for (int k = 0; k < Kd; k++) s += src[(size_t)k * U];
  out[tid] = s * (1.f / Kd);
}

__global__ __launch_bounds__(256) void kAMeanL(const float* __restrict__ A,
                                               float* __restrict__ out) {
  int tid = blockIdx.x * 256 + threadIdx.x;          // Br*C*KU
  int r = tid % KU; size_t bc = tid / KU;
  const float* src = A + bc * S + r;
  float s = 0.f;
#pragma unroll 8
  for (int l = 0; l < L; l++) s += src[(size_t)l * KU];
  out[tid] = s * (1.f / L);
}

// ------------------------------------------------ layer-0 bias (H terms only)
__global__ __launch_bounds__(256) void kBiasH(float* __restrict__ out,
                                              const float* __restrict__ Hm,
                                              const float* __restrict__ PH, int P) {
  int tid = blockIdx.x * 256 + threadIdx.x;          // Br*C*P
  int p = tid % P; int o = (tid / P) % C; int b = tid / (C * P);
  float h0 = Hm[(size_t)(b * 2 + 0) * P + p];
  float h1 = Hm[(size_t)(b * 2 + 1) * P + p];
  out[tid] = 0.1f * (PH[o * 2 + 0] * h0 + PH[o * 2 + 1] * h1);
}

// ------------------------------------------------ layer 0: relu + BN stats
__global__ __launch_bounds__(256) void kLayer0(const float* __restrict__ blk,
                                               const float* __restrict__ blu,
                                               const float* __restrict__ bku,
                                               float* __restrict__ A,
                                               float* __restrict__ stats) {
  size_t tid = (size_t)blockIdx.x * 256 + threadIdx.x;
  int s = (int)(tid % S); size_t bc = tid / S;
  int o = (int)(bc % C);
  int l = s / KU; int r = s % KU; int k = r / U; int u = r % U;
  float v = blk[bc * LK + l * Kd + k] + blu[bc * LU + l * U + u] + bku[bc * KU + r];
  v = fmaxf(v, 0.f);
  A[tid] = v;
  float s1 = v, s2 = v * v;
  for (int m = 16; m > 0; m >>= 1) {
    s1 += __shfl_xor(s1, m, 32);
    s2 += __shfl_xor(s2, m, 32);
  }
  __shared__ float r1[8], r2[8];
  int lane = threadIdx.x & 31, wv = threadIdx.x >> 5;
  if (lane == 0) { r1[wv] = s1; r2[wv] = s2; }
  __syncthreads();
  if (threadIdx.x == 0) {
    float a = 0.f, b = 0.f;
    for (int i = 0; i < 8; i++) { a += r1[i]; b += r2[i]; }
    atomicAdd(&stats[o], a);
    atomicAdd(&stats[C + o], b);
  }
}

// ------------------------------------------------ BN finalize -> (s, t)
__global__ void kBNFin(const float* __restrict__ stats, const float* __restrict__ g,
                       const float* __restrict__ be, float* __restrict__ st, float N) {
  int c = threadIdx.x;
  float m = stats[c] / N;
  float v = stats[C + c] / N - m * m;
  float s = g[c] / sqrtf(v + EPS);
  st[c] = s;
  st[C + c] = be[c] - m * s;
}

// ------------------------------------------------ fold 4 weight matrices (hidden)
__global__ void kFoldLayer(const float* __restrict__ PhA, const float* __restrict__ st,
                           float* __restrict__ W, float* __restrict__ cvec) {
  int o = blockIdx.x, i = threadIdx.x;
  __shared__ float red[64];
  float s = st[i], t = st[C + i];
  for (int m = 0; m < 4; m++) {
    float p = PhA[(m * 64 + o) * 64 + i];
    W[m * 4096 + o * 64 + i] = 2.f * p * s;
    red[i] = 2.f * p * t;
    __syncthreads();
    for (int off = 32; off > 0; off >>= 1) {
      if (i < off) red[i] += red[i + off];
      __syncthreads();
    }
    if (i == 0) cvec[m * 64 + o] = red[0];
    __syncthreads();
  }
}

// ------------------------------------------------ layer-1 broadcast biases (WMMA)
// out[(b,o,p)] = c[o] + sum_i W[o,i]*Am[(b,i,p)] + 0.1*(PH[o,0]*Hm0 + PH[o,1]*Hm1)
// Same LDS tiling as the main GEMM: each mean tile is staged once and reused by
// all 64 output channels (cuts L2 traffic ~64x vs the scalar version).
__global__ __launch_bounds__(128) void kBiasW(float* __restrict__ out,
                                              const float* __restrict__ Am,
                                              const float* __restrict__ Hm,
                                              const float* __restrict__ W,
                                              const float* __restrict__ cvec,
                                              const float* __restrict__ PH, int P) {
  __shared__ float Wsh[64 * 65];
  __shared__ float Ash[4][64][16];
  __shared__ float csh[64], phsh[128];
  int tid = threadIdx.x;
  int wv = tid >> 5, lane = tid & 31;
  int tpb = P >> 6;                      // 64-column tiles per sample
  int b = blockIdx.x / tpb;
  int tile = blockIdx.x - b * tpb;
  int colbase = tile * 64 + wv * 16;

  for (int idx = tid; idx < 4096; idx += 128)
    Wsh[(idx >> 6) * 65 + (idx & 63)] = W[idx];
  if (tid < 64) csh[tid] = cvec[tid];
  phsh[tid] = PH[tid];                   // 128 entries, 128 threads

  const float* srcb = Am + (size_t)b * C * P + colbase;
#pragma unroll
  for (int rr = 0; rr < 2; rr++) {
    int row = lane * 2 + rr;
    const float4* p4 = (const float4*)(srcb + (size_t)row * P);
    float4* d4 = (float4*)&Ash[wv][row][0];
    d4[0] = p4[0]; d4[1] = p4[1]; d4[2] = p4[2]; d4[3] = p4[3];
  }
  __syncthreads();

  int m16 = lane & 15;
  int kb = (lane >> 4) << 1;
  v8f acc[4] = {};
#pragma unroll
  for (int k0 = 0; k0 < 64; k0 += 4) {
    v2f bb;
    bb.x = Ash[wv][k0 + kb][m16];
    bb.y = Ash[wv][k0 + kb + 1][m16];
#pragma unroll
    for (int t = 0; t < 4; t++) {
      const float* wrow = &Wsh[(t * 16 + m16) * 65 + k0 + kb];
      v2f aa; aa.x = wrow[0]; aa.y = wrow[1];
      acc[t] = __builtin_amdgcn_wmma_f32_16x16x4_f32(
          false, aa, false, bb, (short)0, acc[t], false, false);
    }
  }

  int scol = colbase + m16;
  float h0 = 0.1f * Hm[(size_t)(b * 2 + 0) * P + scol];
  float h1 = 0.1f * Hm[(size_t)(b * 2 + 1) * P + scol];
  size_t obase = (size_t)b * C * P + scol;
  int ohalf = (lane < 16) ? 0 : 8;
#pragma unroll
  for (int t = 0; t < 4; t++) {
#pragma unroll
    for (int j = 0; j < 8; j++) {
      int o = t * 16 + j + ohalf;
      out[obase + (size_t)o * P] =
          acc[t][j] + csh[o] + phsh[o * 2] * h0 + phsh[o * 2 + 1] * h1;
    }
  }
}

// ------------------------------------------------ layer-1 main WMMA GEMM
// out[o,s] = relu( sum_i W[o,i]*A[i,s] + c[o] + bias_lk + bias_lu + bias_ku )
// In-place over A.  One block = 4 waves, each wave owns 16 spatial columns.
__global__ __launch_bounds__(128) void kGemm1(float* __restrict__ A,
                                              const float* __restrict__ W,
                                              const float* __restrict__ cvec,
                                              const float* __restrict__ blk,
                                              const float* __restrict__ blu,
                                              const float* __restrict__ bku,
                                              float* __restrict__ stats) {
  __shared__ float Wsh[64 * 65];        // padded rows: no bank conflicts
  __shared__ float Ash[4][64][16];      // per-wave activation tile (i x n)
  __shared__ float lst[2 * C];
  int tid = threadIdx.x;
  int wv = tid >> 5, lane = tid & 31;
  int b = blockIdx.x >> 8;              // 256 column-groups of 64 per sample
  int tile = blockIdx.x & 255;
  int colbase = tile * 64 + wv * 16;

  lst[tid] = 0.f;                       // 128 entries, 128 threads

  for (int idx = tid; idx < 4096; idx += 128)
    Wsh[(idx >> 6) * 65 + (idx & 63)] = W[idx];

  const float* srcb = A + (size_t)b * C * S + colbase;
#pragma unroll
  for (int rr = 0; rr < 2; rr++) {
    int row = lane * 2 + rr;
    const float4* p4 = (const float4*)(srcb + (size_t)row * S);
    float4* d4 = (float4*)&Ash[wv][row][0];
    d4[0] = p4[0]; d4[1] = p4[1]; d4[2] = p4[2]; d4[3] = p4[3];
  }
  __syncthreads();

  int m16 = lane & 15;
  int kb = (lane >> 4) << 1;            // f32 16x16x4 operand K offset per half
  v8f acc[4] = {};

#pragma unroll
  for (int k0 = 0; k0 < 64; k0 += 4) {
    v2f bb;                             // B-matrix 4x16: col = lane%16
    bb.x = Ash[wv][k0 + kb][m16];
    bb.y = Ash[wv][k0 + kb + 1][m16];
#pragma unroll
    for (int t = 0; t < 4; t++) {       // A-matrix 16x4: row = lane%16
      const float* wrow = &Wsh[(t * 16 + m16) * 65 + k0 + kb];
      v2f aa; aa.x = wrow[0]; aa.y = wrow[1];
      acc[t] = __builtin_amdgcn_wmma_f32_16x16x4_f32(
          false, aa, false, bb, (short)0, acc[t], false, false);
    }
  }

  // ---- epilogue: bias add + relu + store + BN stats
  int scol = colbase + m16;
  int l = scol / KU; int r = scol % KU; int k = r / U; int u = r % U;
  size_t bC = (size_t)b * C;
  size_t base_lk = bC * LK + (size_t)(l * Kd + k);
  size_t base_lu = bC * LU + (size_t)(l * U + u);
  size_t base_ku = bC * KU + (size_t)r;
  size_t base_out = bC * S + (size_t)scol;
  int ohalf = (lane < 16) ? 0 : 8;
#pragma unroll
  for (int t = 0; t < 4; t++) {
#pragma unroll
    for (int j = 0; j < 8; j++) {
      int o = t * 16 + j + ohalf;
      float v = acc[t][j] + cvec[o] + blk[base_lk + (size_t)o * LK] +
                blu[base_lu + (size_t)o * LU] + bku[base_ku + (size_t)o * KU];
      v = fmaxf(v, 0.f);
      A[base_out + (size_t)o * S] = v;
      float s1 = v, s2 = v * v;
      s1 += __shfl_xor(s1, 1, 32); s2 += __shfl_xor(s2, 1, 32);
      s1 += __shfl_xor(s1, 2, 32); s2 += __shfl_xor(s2, 2, 32);
      s1 += __shfl_xor(s1, 4, 32); s2 += __shfl_xor(s2, 4, 32);
      s1 += __shfl_xor(s1, 8, 32); s2 += __shfl_xor(s2, 8, 32);
      if (m16 == 0) {
        atomicAdd(&lst[o], s1);
        atomicAdd(&lst[C + o], s2);
      }
    }
  }
  __syncthreads();
  atomicAdd(&stats[tid], lst[tid]);     // 128 entries flushed by 128 threads
}

// ------------------------------------------------ fold output weights (2ch)
__global__ void kFoldFinal(const float* __restrict__ PoA, const float* __restrict__ st,
                           float* __restrict__ W, float* __restrict__ cvec) {
  int mc = blockIdx.x;                  // m*2+co  (8 blocks)
  int i = threadIdx.x;                  // 64
  __shared__ float red[64];
  float p = PoA[mc * 64 + i];
  W[mc * 64 + i] = 2.f * p * st[i];
  red[i] = 2.f * p * st[C + i];
  __syncthreads();
  for (int off = 32; off > 0; off >>= 1) {
    if (i < off) red[i] += red[i + off];
    __syncthreads();
  }
  if (i == 0) cvec[mc] = red[0];
}

// ------------------------------------------------ final broadcast biases (2ch)
__global__ __launch_bounds__(256) void kBiasF(float* __restrict__ out,
                                              const float* __restrict__ Am,
                                              const float* __restrict__ Hm,
                                              const float* __restrict__ W,
                                              const float* __restrict__ cvec,
                                              const float* __restrict__ PH, int P) {
  int tid = blockIdx.x * 256 + threadIdx.x;          // Br*2*P
  int p = tid % P; int co = (tid / P) & 1; int b = tid / (2 * P);
  float acc = cvec[co];
  const float* wr = W + co * 64;
  const float* am = Am + (size_t)b * C * P + p;
#pragma unroll 8
  for (int i = 0; i < C; i++) acc += wr[i] * am[(size_t)i * P];
  acc += 0.1f * (PH[co * 2 + 0] * Hm[(size_t)(b * 2 + 0) * P + p] +
                 PH[co * 2 + 1] * Hm[(size_t)(b * 2 + 1) * P + p]);
  out[tid] = acc;
}

// ------------------------------------------------ final projection + normalize
__global__ __launch_bounds__(256) void kFinal(const float* __restrict__ A,
                                              const float* __restrict__ W,
                                              const float* __restrict__ cvec,
                                              const float* __restrict__ flk,
                                              const float* __restrict__ flu,
                                              const float* __restrict__ fku,
                                              float* __restrict__ out) {
  int bl = blockIdx.x; int b = bl >> 6; int l = bl & 63;
  int t = threadIdx.x;                  // covers KU = 256
  int k = t >> 3; int u = t & 7;
  float f1 = cvec[0], f2 = cvec[1];
  const float* w1 = W;                  // (m=0, co=0)
  const float* w2 = W + 64;             // (m=0, co=1)
  const float* ab = A + (size_t)b * C * S + (size_t)l * KU + t;
#pragma unroll 8
  for (int i = 0; i < C; i++) {
    float a = ab[(size_t)i * S];
    f1 += w1[i] * a;
    f2 += w2[i] * a;
  }
  size_t b2 = (size_t)b * 2;
  f1 += flk[(b2 + 0) * LK + l * Kd + k] + flu[(b2 + 0) * LU + l * U + u] +
        fku[(b2 + 0) * KU + t];
  f2 += flk[(b2 + 1) * LK + l * Kd + k] + flu[(b2 + 1) * LU + l * U + u] +
        fku[(b2 + 1) * KU + t];
  float ss = f1 * f1 + f2 * f2;
  ss += __shfl_xor(ss, 1, 32);
  ss += __shfl_xor(ss, 2, 32);
  ss += __shfl_xor(ss, 4, 32);          // sum over the 8 u-lanes
  float inv = 1.f / sqrtf(ss);
  size_t oi = (size_t)b * S + (size_t)l * KU + t;
  out[2 * oi] = f1 * inv;               // complex64: interleaved (re, im)
  out[2 * oi + 1] = f2 * inv;
}

// =================================================================== driver
extern "C" void kernel_launch(void* const* d_in, const int* in_sizes, int n_in,
                              void* d_out, int out_size, void* d_ws, size_t ws_size,
                              hipStream_t stream) {
  (void)in_sizes; (void)n_in; (void)out_size; (void)ws_size;
  const float* Hre = (const float*)d_in[0];
  const float* Him = (const float*)d_in[1];
  const float* PhA = (const float*)d_in[2];   // (2,4,64,64)
  const float* PhH = (const float*)d_in[3];   // (2,3,64,2)
  const float* g   = (const float*)d_in[4];   // (2,64)
  const float* be  = (const float*)d_in[5];   // (2,64)
  const float* PoA = (const float*)d_in[6];   // (4,2,64)
  const float* PoH = (const float*)d_in[7];   // (3,2,2)

  float* ws = (float*)d_ws;
  size_t off = 0;
  float* A   = ws + off; off += (size_t)Br * C * S;     // activations (in-place both layers)
  float* Hu  = ws + off; off += (size_t)Br * 2 * LK;
  float* Hk  = ws + off; off += (size_t)Br * 2 * LU;
  float* Hl  = ws + off; off += (size_t)Br * 2 * KU;
  float* Au  = ws + off; off += (size_t)Br * C * LK;
  float* Ak  = ws + off; off += (size_t)Br * C * LU;
  float* Al  = ws + off; off += (size_t)Br * C * KU;
  float* Blk = ws + off; off += (size_t)Br * C * LK;
  float* Blu = ws + off; off += (size_t)Br * C * LU;
  float* Bku = ws + off; off += (size_t)Br * C * KU;
  float* Flk = ws + off; off += (size_t)Br * 2 * LK;
  float* Flu = ws + off; off += (size_t)Br * 2 * LU;
  float* Fku = ws + off; off += (size_t)Br * 2 * KU;
  float* stats = ws + off; off += 128;
  float* st0 = ws + off; off += 128;
  float* st1 = ws + off; off += 128;
  float* Wf  = ws + off; off += 4 * 64 * 64;
  float* Cf  = ws + off; off += 4 * 64;
  float* Wo  = ws + off; off += 8 * 64;
  float* Co  = ws + off; off += 8;

  const float Nf = (float)((size_t)Br * S);

  // H axis-means (shared by all layers)
  kHMeanU<<<Br * 2 * LK / 256, 256, 0, stream>>>(Hre, Him, Hu);
  kHMeanK<<<Br * 2 * LU / 256, 256, 0, stream>>>(Hre, Him, Hk);
  kHMeanL<<<Br * 2 * KU / 256, 256, 0, stream>>>(Hre, Him, Hl);

  // ---- layer 0 (A == 0): only H-bias terms, relu, BN stats
  kBiasH<<<Br * C * LK / 256, 256, 0, stream>>>(Blk, Hu, PhH + 0 * 128, LK);
  kBiasH<<<Br * C * LU / 256, 256, 0, stream>>>(Blu, Hk, PhH + 1 * 128, LU);
  kBiasH<<<Br * C * KU / 256, 256, 0, stream>>>(Bku, Hl, PhH + 2 * 128, KU);
  kZeroStats<<<1, 128, 0, stream>>>(stats);
  kLayer0<<<(int)(((size_t)Br * C * S) / 256), 256, 0, stream>>>(Blk, Blu, Bku, A, stats);
  kBNFin<<<1, 64, 0, stream>>>(stats, g, be, st0, Nf);

  // ---- layer 1: fold BN affine into weights, means, WMMA biases, WMMA GEMM
  kAMeanU<<<Br * C * LK / 256, 256, 0, stream>>>(A, Au);
  kAMeanK<<<Br * C * LU / 256, 256, 0, stream>>>(A, Ak);
  kAMeanL<<<Br * C * KU / 256, 256, 0, stream>>>(A, Al);
  kFoldLayer<<<64, 64, 0, stream>>>(PhA + 4 * 64 * 64, st0, Wf, Cf);
  kBiasW<<<Br * (LK / 64), 128, 0, stream>>>(Blk, Au, Hu, Wf + 1 * 4096, Cf + 64,  PhH + (3 + 0) * 128, LK);
  kBiasW<<<Br * (LU / 64), 128, 0, stream>>>(Blu, Ak, Hk, Wf + 2 * 4096, Cf + 128, PhH + (3 + 1) * 128, LU);
  kBiasW<<<Br * (KU / 64), 128, 0, stream>>>(Bku, Al, Hl, Wf + 3 * 4096, Cf + 192, PhH + (3 + 2) * 128, KU);
  kZeroStats<<<1, 128, 0, stream>>>(stats);
  kGemm1<<<Br * 256, 128, 0, stream>>>(A, Wf, Cf, Blk, Blu, Bku, stats);
  kBNFin<<<1, 64, 0, stream>>>(stats, g + 64, be + 64, st1, Nf);

  // ---- output layer (2 channels) + u-axis L2 normalization
  kAMeanU<<<Br * C * LK / 256, 256, 0, stream>>>(A, Au);
  kAMeanK<<<Br * C * LU / 256, 256, 0, stream>>>(A, Ak);
  kAMeanL<<<Br * C * KU / 256, 256, 0, stream>>>(A, Al);
  kFoldFinal<<<8, 64, 0, stream>>>(PoA, st1, Wo, Co);
  kBiasF<<<Br * 2 * LK / 256, 256, 0, stream>>>(Flk, Au, Hu, Wo + 1 * 128, Co + 2, PoH + 0 * 4, LK);
  kBiasF<<<Br * 2 * LU / 256, 256, 0, stream>>>(Flu, Ak, Hk, Wo + 2 * 128, Co + 4, PoH + 1 * 4, LU);
  kBiasF<<<Br * 2 * KU / 256, 256, 0, stream>>>(Fku, Al, Hl, Wo + 3 * 128, Co + 6, PoH + 2 * 4, KU);
  kFinal<<<Br * L, 256, 0, stream>>>(A, Wo, Co, Flk, Flu, Fku, (float*)d_out);
}